// VirtualParameter_9354438771003
// MI455X (gfx1250) — compile-verified
//
#include <hip/hip_runtime.h>
#include <stdint.h>

// ---------------------------------------------------------------------------
// VirtualParameter weighted bank gather for MI455X (gfx1250).
//   out[b, p] = sum_s prob[b,s] * param[p, idx[b,s]]     p in [0, 512*2048)
// Memory-bound: 256 MB param read once + 256 MB out written once (~22 us at
// 23.3 TB/s). Data path: TDM tensor_load_to_lds (64 KB contiguous tile,
// HW-padded rows for conflict-free LDS banking) -> VALU weighted sum ->
// non-temporal streaming stores.
// ---------------------------------------------------------------------------

typedef unsigned int u32x4 __attribute__((ext_vector_type(4)));
typedef int          i32x8 __attribute__((ext_vector_type(8)));
typedef int          i32x4 __attribute__((ext_vector_type(4)));

#define BANK      64
#define ROW_PAD   65            // 64 data dwords + 1 pad dword (TDM pad feature)
#define BATCH     64
#define SELECTED  4
#define POS_TILE  256           // positions per workgroup (tile = 64 KB contiguous)
#define NPOS      (512LL * 2048LL)

__global__ __launch_bounds__(POS_TILE)
void vp_weighted_gather(const float* __restrict__ param,
                        const int*   __restrict__ sel_idx,
                        const float* __restrict__ sel_prob,
                        float*       __restrict__ out)
{
    __shared__ float tile[POS_TILE * ROW_PAD];     // 66,560 B, rows padded by TDM
    __shared__ int   s_idx [BATCH * SELECTED];     // 1 KB
    __shared__ float s_prob[BATCH * SELECTED];     // 1 KB

    const int tid = threadIdx.x;

    // Stage the tiny selection tables (exactly 256 entries each).
    s_idx[tid]  = sel_idx[tid];
    s_prob[tid] = sel_prob[tid];

    const unsigned long long tile_pos = (unsigned long long)blockIdx.x * POS_TILE;

    // ---- TDM: DMA 256 rows x 64 dwords from global into LDS, padding each
    // ---- 64-dword row with 1 dword so lane t reads bank (t + idx) % 64.
    if (tid < 32) {  // one wave issues the descriptor (EXEC ignored by TDM)
        const unsigned lds_addr = (unsigned)(uintptr_t)(void*)tile; // LDS byte offset (addr[31:0])
        const unsigned long long gaddr =
            (unsigned long long)(uintptr_t)param + tile_pos * (BANK * 4ull);

        // D# group 0 (128b): count=1 | lds_addr | global_addr[56:0] | type=2
        u32x4 g0;
        g0[0] = 1u;                                              // count=1, user mode
        g0[1] = lds_addr;                                        // lds_addr[31:0]
        g0[2] = (unsigned)(gaddr & 0xFFFFFFFFull);               // global_addr[31:0]
        g0[3] = (unsigned)((gaddr >> 32) & 0x1FFFFFFull)         // global_addr[56:32]
              | (2u << 30);                                      // type = 2 ("image")

        // D# group 1 (256b)
        const unsigned data_size    = 2;        // 4-byte elements
        const unsigned pad_enable   = 1;
        const unsigned pad_interval = 5;        // pad every 2^(5+1) = 64 dwords
        const unsigned pad_amount   = 0;        // insert 1 dword
        const unsigned tensor_dim0  = BANK;     // 64 elements per row
        const unsigned tensor_dim1  = (unsigned)NPOS;
        const unsigned tile_dim0    = BANK;     // 64
        const unsigned tile_dim1    = POS_TILE; // 256 rows
        const unsigned tile_dim2    = 1;
        const unsigned dim0_stride  = BANK;     // contiguous rows

        i32x8 g1;
        g1[0] = (int)((data_size    << 16) |
                      (pad_enable   << 20) |
                      (pad_interval << 22) |
                      (pad_amount   << 25));                     // wg_mask=0 (no cluster)
        g1[1] = (int)((tensor_dim0 & 0xFFFFu) << 16);            // barrier_addr=0 | dim0 lo16
        g1[2] = (int)((tensor_dim0 >> 16) |
                      ((tensor_dim1 & 0xFFFFu) << 16));
        g1[3] = (int)((tensor_dim1 >> 16) | (tile_dim0 << 16));
        g1[4] = (int)(tile_dim1 | (tile_dim2 << 16));
        g1[5] = (int)dim0_stride;                                // dim0_stride[31:0]
        g1[6] = 0;                                               // dim0_stride[47:32] | dim1_stride lo16
        g1[7] = 0;                                               // dim1_stride[47:16]

        // D# groups 2/3: higher dims collapsed to size 1.
        i32x4 g2;
        g2[0] = 1;                 // tensor_dim2 = 1
        g2[1] = 1;                 // tensor_dim3 = 1
        g2[2] = 0;                 // tensor_dim2_stride lo
        g2[3] = (int)(1u << 16);   // tile_dim3 = 1

        i32x4 g3;
        g3[0] = 0;                 // tensor_dim3_stride lo
        g3[1] = (int)(1u << 16);   // tensor_dim4 = 1
        g3[2] = (int)(1u << 16);   // tile_dim4 = 1
        g3[3] = 0;

        i32x8 g4 = {0, 0, 0, 0, 0, 0, 0, 0};   // uncharacterized 5th operand: zero-fill

        __builtin_amdgcn_tensor_load_to_lds(g0, g1, g2, g3, g4, 0);
        __builtin_amdgcn_s_wait_tensorcnt(0);
    }
    __syncthreads();

    // ---- Compute: thread owns one position; conflict-free LDS gathers.
    const float* __restrict__ prow = &tile[tid * ROW_PAD];
    const unsigned long long  opos = tile_pos + (unsigned)tid;

    #pragma unroll 4
    for (int b = 0; b < BATCH; ++b) {
        const int   i0 = s_idx [b * SELECTED + 0];
        const int   i1 = s_idx [b * SELECTED + 1];
        const int   i2 = s_idx [b * SELECTED + 2];
        const int   i3 = s_idx [b * SELECTED + 3];
        const float p0 = s_prob[b * SELECTED + 0];
        const float p1 = s_prob[b * SELECTED + 1];
        const float p2 = s_prob[b * SELECTED + 2];
        const float p3 = s_prob[b * SELECTED + 3];

        float acc = p0 * prow[i0];
        acc = fmaf(p1, prow[i1], acc);
        acc = fmaf(p2, prow[i2], acc);
        acc = fmaf(p3, prow[i3], acc);

        // Streaming write-once output: non-temporal, bypass cache residency.
        __builtin_nontemporal_store(acc, &out[(unsigned long long)b * NPOS + opos]);
    }
}

extern "C" void kernel_launch(void* const* d_in, const int* in_sizes, int n_in,
                              void* d_out, int out_size, void* d_ws, size_t ws_size,
                              hipStream_t stream) {
    (void)in_sizes; (void)n_in; (void)out_size; (void)d_ws; (void)ws_size;
    const float* param    = (const float*)d_in[0];
    const int*   sel_idx  = (const int*)  d_in[1];
    const float* sel_prob = (const float*)d_in[2];
    float*       out      = (float*)d_out;

    const int blocks = (int)(NPOS / POS_TILE);   // 4096
    vp_weighted_gather<<<blocks, POS_TILE, 0, stream>>>(param, sel_idx, sel_prob, out);
}